// ProgressiveTransfer_32134945308759
// MI455X (gfx1250) — compile-verified
//
#include <hip/hip_runtime.h>
#include <math.h>

// Progressive color-transfer patch statistics:
//   7x7 clipped-window mean/std for guidance & source -> alpha/beta planes.
// Memory-bound (~300MB traffic -> ~13us floor at 23.3 TB/s). Fused single
// kernel: async-to-LDS halo staging (ASYNCcnt path), VALU horizontal 7-tap,
// vertical 7-tap as banded-matrix product on the CDNA5 matrix pipe via
// V_WMMA_F32_16X16X4_F32 (K=24 in six K=4 steps), non-temporal output stores.

#define IMG_H 2048
#define IMG_W 3072
#define IMG_C 3
#define EPS_F 0.002f

#define BX 64            // output tile width per block (4 waves in x)
#define BY 32            // output tile height per block (2 waves in y)
#define RAW_R 38         // BY + 6 halo rows
#define RAW_C 70         // BX + 6 halo cols
#define RAW_P 72         // raw LDS pitch (floats)
#define HS_R 40          // 38 used + 2 zero rows (WMMA K-loop tail reads)
#define HS_P 72          // hsum pitch: 2*72 % 64 = 16 -> lane halves hit disjoint banks

typedef __attribute__((ext_vector_type(2))) float v2f;
typedef __attribute__((ext_vector_type(8))) float v8f;

#if defined(__has_builtin)
#if __has_builtin(__builtin_amdgcn_global_load_async_to_lds_b32)
#define HAVE_ASYNC_LDS 1
#endif
#endif
#ifndef HAVE_ASYNC_LDS
#define HAVE_ASYNC_LDS 0
#endif

#if HAVE_ASYNC_LDS
typedef __attribute__((address_space(1))) int gas_i32;   // global (AS1) 32-bit
typedef __attribute__((address_space(3))) int las_i32;   // LDS (AS3) 32-bit
#endif

static __device__ __forceinline__ void wait_async_then_barrier() {
#if HAVE_ASYNC_LDS
#if __has_builtin(__builtin_amdgcn_s_wait_asynccnt)
    __builtin_amdgcn_s_wait_asynccnt(0);
#else
    asm volatile("s_wait_asynccnt 0" ::: "memory");
#endif
#endif
    __syncthreads();
}

__global__ __launch_bounds__(256)
void ProgressiveTransfer_32134945308759_kernel(const float* __restrict__ gsrc,
                                               const float* __restrict__ ssrc,
                                               float* __restrict__ out) {
    __shared__ float raw[RAW_R * RAW_P];        // one image's halo tile at a time
    __shared__ float hs[4][HS_R * HS_P];        // s1_g, s2_g, s1_s, s2_s

    const int tid = threadIdx.x;
    const int xb = blockIdx.x * BX;
    const int yb = blockIdx.y * BY;
    const int ch = blockIdx.z;

    const float* srcs[2] = { gsrc + (size_t)ch * IMG_H * IMG_W,
                             ssrc + (size_t)ch * IMG_H * IMG_W };

    // ---- Stage halo + horizontal 7-tap pass (per image, sequential to fit LDS)
    for (int im = 0; im < 2; ++im) {
        const float* __restrict__ src = srcs[im];
        for (int idx = tid; idx < RAW_R * RAW_C; idx += 256) {
            int r = idx / RAW_C, c = idx - r * RAW_C;
            int gy = yb + r - 3, gx = xb + c - 3;
            int loff = r * RAW_P + c;
            bool inb = (gy >= 0) && (gy < IMG_H) && (gx >= 0) && (gx < IMG_W);
#if HAVE_ASYNC_LDS
            if (inb) {
                // GLOBAL_LOAD_ASYNC_TO_LDS_B32: memory -> LDS, no VGPR round-trip,
                // tracked by ASYNCcnt. OOB lanes fall through to the ds_store below
                // (disjoint LDS addresses -> no ordering hazard).
                __builtin_amdgcn_global_load_async_to_lds_b32(
                    (gas_i32*)(src + (size_t)gy * IMG_W + gx),
                    (las_i32*)&raw[loff], 0, 0);
            } else {
                raw[loff] = 0.0f;
            }
#else
            float v = 0.0f;
            if (inb) v = src[(size_t)gy * IMG_W + gx];
            raw[loff] = v;
#endif
        }
        wait_async_then_barrier();
        // hsum planes: rows 0..37 real, rows 38..39 forced zero (WMMA K tail)
        for (int idx = tid; idx < HS_R * BX; idx += 256) {
            int r = idx >> 6, x = idx & 63;
            float s1 = 0.0f, s2 = 0.0f;
            if (r < RAW_R) {
                const float* row = &raw[r * RAW_P + x];
#pragma unroll
                for (int d = 0; d < 7; ++d) { float v = row[d]; s1 += v; s2 += v * v; }
            }
            hs[2 * im + 0][r * HS_P + x] = s1;
            hs[2 * im + 1][r * HS_P + x] = s2;
        }
        __syncthreads();
    }

    // ---- Vertical 7-tap pass as banded 16x24 x 24x16 product on the matrix pipe.
    // Wave w owns a 16x16 output tile: tx = w&3 (x), ty = w>>2 (y).
    const int wave = tid >> 5;          // wave32
    const int lane = tid & 31;
    const int tx = wave & 3, ty = wave >> 2;
    const int half = lane >> 4;         // K high bit for A/B operands
    const int mn = lane & 15;           // M for A, N for B/D
    const int rbase = 16 * ty;          // hsum row origin of this subtile's halo
    const int cbase = 16 * tx;

    v8f acc0 = {}, acc1 = {}, acc2 = {}, acc3 = {};
#pragma unroll
    for (int t = 0; t < 6; ++t) {
        const int k0 = 4 * t + 2 * half;            // local K of this lane's VGPR0
        // Band[m][k] = 1 iff output row m's window covers halo row k (k in [m, m+6]).
        v2f a;
        a.x = (k0     >= mn && k0     <= mn + 6) ? 1.0f : 0.0f;
        a.y = (k0 + 1 >= mn && k0 + 1 <= mn + 6) ? 1.0f : 0.0f;
        const int row0 = (rbase + k0) * HS_P + cbase + mn;
        const int row1 = row0 + HS_P;
        v2f b;
        b.x = hs[0][row0]; b.y = hs[0][row1];
        acc0 = __builtin_amdgcn_wmma_f32_16x16x4_f32(false, a, false, b, (short)0, acc0, false, false);
        b.x = hs[1][row0]; b.y = hs[1][row1];
        acc1 = __builtin_amdgcn_wmma_f32_16x16x4_f32(false, a, false, b, (short)0, acc1, false, false);
        b.x = hs[2][row0]; b.y = hs[2][row1];
        acc2 = __builtin_amdgcn_wmma_f32_16x16x4_f32(false, a, false, b, (short)0, acc2, false, false);
        b.x = hs[3][row0]; b.y = hs[3][row1];
        acc3 = __builtin_amdgcn_wmma_f32_16x16x4_f32(false, a, false, b, (short)0, acc3, false, false);
    }

    // ---- Epilogue: clipped-count stats, alpha/beta, streaming (NT) stores.
    // D layout: lane -> N = lane&15; VGPR i -> M = i + 8*(lane>>4).
    const int x = xb + cbase + mn;
    const int x0 = (x - 3 < 0) ? 0 : x - 3;
    const int x1 = (x + 3 > IMG_W - 1) ? IMG_W - 1 : x + 3;
    const float cx = (float)(x1 - x0 + 1);
#pragma unroll
    for (int i = 0; i < 8; ++i) {
        const int m = i + 8 * half;
        const int y = yb + rbase + m;
        const int y0 = (y - 3 < 0) ? 0 : y - 3;
        const int y1 = (y + 3 > IMG_H - 1) ? IMG_H - 1 : y + 3;
        const float cnt = cx * (float)(y1 - y0 + 1);
        const float s1g = acc0[i], s2g = acc1[i];
        const float s1s = acc2[i], s2s = acc3[i];
        const float mg = s1g / cnt;
        const float ms = s1s / cnt;
        const float inv = 1.0f / (cnt - 1.0f);
        float vg = (s2g - s1g * mg) * inv;
        float vs = (s2s - s1s * ms) * inv;
        vg = vg > 0.0f ? vg : 0.0f;
        vs = vs > 0.0f ? vs : 0.0f;
        const float alpha = sqrtf(vg) / (sqrtf(vs) + EPS_F);
        const float beta = mg - alpha * ms;
        const size_t off = ((size_t)ch * IMG_H + y) * IMG_W + x;
        __builtin_nontemporal_store(alpha, &out[off]);                               // plane [0][ch]
        __builtin_nontemporal_store(beta, &out[(size_t)IMG_C * IMG_H * IMG_W + off]); // plane [1][ch]
    }
}

extern "C" void kernel_launch(void* const* d_in, const int* in_sizes, int n_in,
                              void* d_out, int out_size, void* d_ws, size_t ws_size,
                              hipStream_t stream) {
    (void)in_sizes; (void)n_in; (void)out_size; (void)d_ws; (void)ws_size;
    const float* guidance = (const float*)d_in[0];
    const float* source   = (const float*)d_in[1];
    float* out = (float*)d_out;
    dim3 grid(IMG_W / BX, IMG_H / BY, IMG_C);   // 48 x 64 x 3
    dim3 block(256);                            // 8 wave32 waves -> 8 16x16 tiles
    ProgressiveTransfer_32134945308759_kernel<<<grid, block, 0, stream>>>(guidance, source, out);
}